// ModelNew_80908593923089
// MI455X (gfx1250) — compile-verified
//
#include <hip/hip_runtime.h>
#include <hip/hip_bf16.h>

typedef __attribute__((ext_vector_type(2))) float v2f;
typedef __attribute__((ext_vector_type(8))) float v8f;

#define WAVES   9
#define THREADS (WAVES * 32)
#define CI_PAD  34   // ci dim padded 32->34: bank-conflict-free, keeps 8B alignment
#define WSLOTS  34   // wslot dim padded 32->34: zero rows for M-tail (ow'=30,31), no predication

// LDS: xs 5*5*34*34*4 = 115600 B, bs 2*64*34*4 = 17408 B, pool 640*4 = 2560 B  (~136 KB)
__global__ __launch_bounds__(THREADS)
void convt_pool_sum_kernel(const float* __restrict__ x,
                           const float* __restrict__ w,     // [ci=32][co=64][5][5][5]
                           const float* __restrict__ bias,  // [64]
                           float* __restrict__ out)         // [16][1][5][10][10]
{
    __shared__ float    xs[5 * 5 * WSLOTS * CI_PAD];  // [islot][jslot][wslot][ci]
    __shared__ float    bs[2][64 * CI_PAD];           // double-buffered [c][ci] weight tile
    __shared__ unsigned pool[10 * 64];                // [wo][c] ordered-uint max keys

    const int tid = threadIdx.x;
    const int wg  = blockIdx.x;        // 800 = 16 n * 5 do * 10 ho
    const int n   = wg / 50;
    const int rem = wg % 50;
    const int dot = rem / 10;          // pooled-output d index (0..4)
    const int hot = rem % 10;          // pooled-output h index (0..9)

    // ---- stage x slab: coalesced global reads (iw fastest), ci-fastest LDS writes ----
    // wslot = iw+1 covers iw in [-1, 30]; wslots 32,33 are zero pad for the M-tail rows.
    for (int idx = tid; idx < 32 * 5 * 5 * WSLOTS; idx += THREADS) {
        int iwm = idx % WSLOTS;        // wslot
        int t1  = idx / WSLOTS;
        int js  = t1 % 5;
        int t2  = t1 / 5;
        int is  = t2 % 5;
        int ci  = t2 / 5;
        int id  = 3 * dot - 1 + is;
        int ih  = 3 * hot - 1 + js;
        int iw  = iwm - 1;
        float v = 0.0f;
        if (id >= 0 && id < 16 && ih >= 0 && ih < 32 && iw >= 0 && iw < 31) {
            v = x[((ci * 16 + id) * 32 + ih) * 32 + iw];
        }
        xs[((is * 5 + js) * WSLOTS + iwm) * CI_PAD + ci] = v;
    }
    for (int idx = tid; idx < 640; idx += THREADS) pool[idx] = 0u;  // 0 < key(-FLT_MAX)

    const int wv   = tid >> 5;         // wave 0..8
    const int lane = tid & 31;
    const int lhi  = lane >> 4;        // 0/1: K-half per ISA A/B layout
    const int lm   = lane & 15;

    float bias_l[4];
#pragma unroll
    for (int nt = 0; nt < 4; ++nt) bias_l[nt] = bias[nt * 16 + lm];

    __syncthreads();

    const int odl = wv / 3;            // od' = 3*do + odl
    const int ohl = wv % 3;            // oh' = 3*ho + ohl
    int tapc = 0;                      // global tap counter -> B buffer parity

    for (int cls = 0; cls < 8; ++cls) {
        const int pd = (cls >> 2) & 1, ph = (cls >> 1) & 1, pw = cls & 1;
        const int td = pd ? 2 : 3, th = ph ? 2 : 3, tw = pw ? 2 : 3;

        // C fragments: y[od',oh', ow' = mt*16+m, c = nt*16+n], init with bias
        v8f acc[2][4];
#pragma unroll
        for (int mt = 0; mt < 2; ++mt)
#pragma unroll
            for (int nt = 0; nt < 4; ++nt) {
                v8f c;
#pragma unroll
                for (int i = 0; i < 8; ++i) c[i] = bias_l[nt];
                acc[mt][nt] = c;
            }

        for (int kd = 0; kd < td; ++kd)
        for (int kh = 0; kh < th; ++kh)
        for (int kw = 0; kw < tw; ++kw) {
            const int buf    = tapc & 1;
            const int tapoff = (2 * kd + pd) * 25 + (2 * kh + ph) * 5 + (2 * kw + pw);

            // stage next weight tile into its buffer, then one barrier (safe: all
            // waves passed the previous barrier only after finishing the last
            // compute that read this buffer)
            for (int e = tid; e < 64 * 32; e += THREADS) {
                int ci = e & 31, c = e >> 5;
                bs[buf][c * CI_PAD + ci] = w[(ci * 64 + c) * 125 + tapoff];
            }
            __syncthreads();

            const int is    = odl + 2 - kd;            // slab id slot (0..4)
            const int js    = ohl + 2 - kh;            // slab ih slot (0..4)
            const int abase = (is * 5 + js) * WSLOTS;

#pragma unroll
            for (int k0 = 0; k0 < 32; k0 += 4) {       // K = 32 ci, 4 per WMMA
                v2f a[2];
#pragma unroll
                for (int mt = 0; mt < 2; ++mt) {
                    int owp = mt * 16 + lm;            // ow' (M index), 30/31 -> zero pad rows
                    int ws  = owp + 2 - kw;            // wslot = (ow'+1-kw') + 1, in [0,33]
                    a[mt] = *(const v2f*)&xs[(abase + ws) * CI_PAD + (k0 + lhi * 2)];
                }
                v2f bf[4];
#pragma unroll
                for (int nt = 0; nt < 4; ++nt) {
                    bf[nt] = *(const v2f*)&bs[buf][(nt * 16 + lm) * CI_PAD + (k0 + lhi * 2)];
                }
#pragma unroll
                for (int mt = 0; mt < 2; ++mt)
#pragma unroll
                    for (int nt = 0; nt < 4; ++nt)
                        acc[mt][nt] = __builtin_amdgcn_wmma_f32_16x16x4_f32(
                            false, a[mt], false, bf[nt],
                            (short)0, acc[mt][nt], false, false);
            }
            ++tapc;
        }

        // ---- fused 6x6x6 max-pool: ordered-uint atomic max into pool[wo][c] ----
#pragma unroll
        for (int mt = 0; mt < 2; ++mt)
#pragma unroll
            for (int nt = 0; nt < 4; ++nt)
#pragma unroll
                for (int v = 0; v < 8; ++v) {
                    int mloc = v + lhi * 8;            // C layout: M = v + 8*(lane/16)
                    int owp  = mt * 16 + mloc;
                    if (owp < 30) {
                        int wo = owp / 3;
                        unsigned u   = __float_as_uint(acc[mt][nt][v]);
                        unsigned key = (u & 0x80000000u) ? ~u : (u | 0x80000000u);
                        atomicMax(&pool[wo * 64 + nt * 16 + lm], key);
                    }
                }
    }

    __syncthreads();

    // ---- channel sum (deterministic serial order), 10 outputs per WG ----
    if (tid < 10) {
        float s = 0.0f;
        for (int c = 0; c < 64; ++c) {
            unsigned key = pool[tid * 64 + c];
            unsigned u   = (key & 0x80000000u) ? (key & 0x7fffffffu) : ~key;
            s += __uint_as_float(u);
        }
        out[n * 500 + dot * 100 + hot * 10 + tid] = s;
    }
}

extern "C" void kernel_launch(void* const* d_in, const int* in_sizes, int n_in,
                              void* d_out, int out_size, void* d_ws, size_t ws_size,
                              hipStream_t stream) {
    (void)in_sizes; (void)n_in; (void)d_ws; (void)ws_size; (void)out_size;
    const float* x    = (const float*)d_in[0];   // [16,32,16,32,32]
    const float* w    = (const float*)d_in[1];   // [32,64,5,5,5]
    const float* bias = (const float*)d_in[2];   // [64]
    float* out        = (float*)d_out;           // [16,1,5,10,10] = 8000

    dim3 grid(16 * 5 * 10);   // one WG per (n, do, ho)
    dim3 block(THREADS);      // 9 waves of 32
    hipLaunchKernelGGL(convt_pool_sum_kernel, grid, block, 0, stream, x, w, bias, out);
}